// DGM_d_17987323036004
// MI455X (gfx1250) — compile-verified
//
#include <hip/hip_runtime.h>

typedef __attribute__((ext_vector_type(2))) float v2f;
typedef __attribute__((ext_vector_type(8))) float v8f;

#define NPTS 16384
#define DIM  64
#define KNN  20
#define ND   (NPTS * DIM)
#define NK   (NPTS * KNN)

// ---------------------------------------------------------------------------
// Precompute squared norms sq[j] = sum_d x[j][d]^2 into workspace.
// ---------------------------------------------------------------------------
__global__ void sq_kernel(const float* __restrict__ x, float* __restrict__ sq) {
    int j = blockIdx.x * blockDim.x + threadIdx.x;
    if (j >= NPTS) return;
    const float4* row = (const float4*)(x + (size_t)j * DIM);
    float s = 0.f;
#pragma unroll
    for (int q = 0; q < DIM / 4; ++q) {
        float4 v = row[q];
        s += v.x * v.x + v.y * v.y + v.z * v.z + v.w * v.w;
    }
    sq[j] = s;
}

// ---------------------------------------------------------------------------
// KNN kernel: each wave owns a 16-row tile. Gram tile via fp32 WMMA
// (V_WMMA_F32_16X16X4_F32, K=64 as 16 chained k-chunks), top-20 selection
// via per-row insertion lists in LDS, exact d2 recompute for logprobs.
// ---------------------------------------------------------------------------
__launch_bounds__(128)
__global__ void knn_kernel(const float* __restrict__ x,
                           const float* __restrict__ sq,
                           const float* __restrict__ temp,
                           float* __restrict__ out) {
    __shared__ float s_dot[4][16][16];
    __shared__ float s_bd[4][16][KNN];
    __shared__ int   s_bi[4][16][KNN];

    const int tid     = threadIdx.x;
    const int wave    = tid >> 5;
    const int lane    = tid & 31;
    const int lane_lo = lane & 15;
    const int lane_hi = lane >> 4;
    const int rowBase = blockIdx.x * 64 + wave * 16;

    // ---- copy x into out[0 .. ND): this block's 64 rows, coalesced float4 ----
    {
        const float4* src = (const float4*)(x + (size_t)blockIdx.x * 64 * DIM);
        float4* dst       = (float4*)(out + (size_t)blockIdx.x * 64 * DIM);
#pragma unroll
        for (int q = 0; q < (64 * DIM / 4) / 128; ++q)
            dst[q * 128 + tid] = src[q * 128 + tid];
    }

    // ---- init per-row top-K lists ----
    if (lane < 16) {
#pragma unroll
        for (int k = 0; k < KNN; ++k) {
            s_bd[wave][lane][k] = 3.4e38f;
            s_bi[wave][lane][k] = 0;
        }
    }

    // ---- load A tile (16 rows x 64 k) into registers, WMMA 16x4 f32 layout:
    //      lane L: M = L%16, K = 4c + 2*(L/16) + {0,1}  -> one b64 load/chunk
    v2f a[16];
    {
        const float* arow = x + (size_t)(rowBase + lane_lo) * DIM + 2 * lane_hi;
#pragma unroll
        for (int c = 0; c < 16; ++c)
            a[c] = *(const v2f*)(arow + 4 * c);
    }

    const float sqi = sq[rowBase + lane_lo];
    float dmax = 3.4e38f;

    for (int n0 = 0; n0 < NPTS; n0 += 16) {
        // B tile (4x16 per chunk): lane L holds N = n0 + L%16, K = 4c + 2*(L/16)+{0,1}
        const float* brow = x + (size_t)(n0 + lane_lo) * DIM + 2 * lane_hi;
        v8f acc = {0.f, 0.f, 0.f, 0.f, 0.f, 0.f, 0.f, 0.f};
#pragma unroll
        for (int c = 0; c < 16; ++c) {
            v2f b = *(const v2f*)(brow + 4 * c);
            acc = __builtin_amdgcn_wmma_f32_16x16x4_f32(
                false, a[c], false, b, (short)0, acc, false, false);
        }

        // scatter 16x16 dot tile to LDS (C layout: VGPR r -> row r + 8*lane_hi)
#pragma unroll
        for (int r = 0; r < 8; ++r)
            s_dot[wave][r + 8 * lane_hi][lane_lo] = acc[r];
        // wave-local LDS visibility: split dep counter, no barrier needed
        asm volatile("s_wait_dscnt 0" ::: "memory");

        if (lane < 16) {
            const int t = lane;
            for (int c = 0; c < 16; ++c) {
                const int n = n0 + c;
                const float d2 = sqi + sq[n] - 2.0f * s_dot[wave][t][c];
                if (d2 < dmax) {  // strict: ties keep earlier (lower) index
                    int pos = KNN - 1;
                    while (pos > 0 && s_bd[wave][t][pos - 1] > d2) {
                        s_bd[wave][t][pos] = s_bd[wave][t][pos - 1];
                        s_bi[wave][t][pos] = s_bi[wave][t][pos - 1];
                        --pos;
                    }
                    s_bd[wave][t][pos] = d2;
                    s_bi[wave][t][pos] = n;
                    dmax = s_bd[wave][t][KNN - 1];
                }
            }
        }
    }

    // ---- finalize: edges + exact-gather logprobs ----
    if (lane < 16) {
        const int i = rowBase + lane;
        float tv = *temp;
        tv = fminf(fmaxf(tv, -5.0f), 5.0f);
        const float scale = __expf(tv);
        const float* xi = x + (size_t)i * DIM;
        for (int k = 0; k < KNN; ++k) {
            const int j = s_bi[wave][lane][k];
            const float* xj = x + (size_t)j * DIM;
            float d2r = 0.f;
#pragma unroll
            for (int d = 0; d < DIM; ++d) {
                const float df = xj[d] - xi[d];
                d2r = fmaf(df, df, d2r);
            }
            const size_t e = (size_t)i * KNN + k;
            out[ND + e]           = (float)j;   // edges[0]: neighbor index
            out[ND + NK + e]      = (float)i;   // edges[1]: row index
            out[ND + 2 * NK + e]  = -d2r * scale;  // logprobs
        }
    }
}

// ---------------------------------------------------------------------------
extern "C" void kernel_launch(void* const* d_in, const int* in_sizes, int n_in,
                              void* d_out, int out_size, void* d_ws, size_t ws_size,
                              hipStream_t stream) {
    const float* x    = (const float*)d_in[0];
    // d_in[1] (A) unused: embed_f is identity
    const float* temp = (const float*)d_in[2];
    float* out = (float*)d_out;
    float* sq  = (float*)d_ws;  // NPTS floats = 64 KB

    sq_kernel<<<NPTS / 256, 256, 0, stream>>>(x, sq);
    knn_kernel<<<NPTS / 64, 128, 0, stream>>>(x, sq, temp, out);
}